// ContextBasedSumAttention_64484638982248
// MI455X (gfx1250) — compile-verified
//
#include <hip/hip_runtime.h>
#include <hip/hip_bf16.h>
#include <math.h>

// Problem constants (from reference)
#define BSZ 32
#define SSZ 2048
#define HSZ 1024

// Flash pass decomposition
#define NCHUNK 8                       // S-chunks per batch -> 32*8 = 256 workgroups
#define ROWS_PER_CHUNK (SSZ / NCHUNK)  // 256 rows
#define THREADS 256                    // 8 waves of 32 (wave32 native)
#define NWAVES (THREADS / 32)
#define ROWS_PER_WAVE (ROWS_PER_CHUNK / NWAVES) // 32
#define PSTRIDE (HSZ + 8)              // per-partial stride in floats ([0]=M,[1]=L,[8..]=acc)

typedef __attribute__((ext_vector_type(2))) float v2f;
typedef __attribute__((ext_vector_type(8))) float v8f;

// ---------------------------------------------------------------------------
// Kernel 1: v[b,h] = sum_k h_t[b,k] * W[k,h]   ([32x1024] @ [1024x1024])
// One wave per 16x16 output tile, V_WMMA_F32_16X16X4_F32 over 256 K-steps.
// A (16x4 f32) layout: lanes 0-15 hold M, VGPR0/1 = K{0,1} (lo half), K{2,3} (hi half).
// B (4x16 f32) layout mirrored: lanes hold N, VGPRs hold K pairs.
// C/D (16x16 f32): VGPR j -> M = j + 8*(lane>>4), N = lane&15.
// ---------------------------------------------------------------------------
__global__ __launch_bounds__(32)
void cbsa_htW_wmma(const float* __restrict__ h_t, const float* __restrict__ W,
                   float* __restrict__ v)
{
    const int lane  = threadIdx.x;          // 0..31
    const int mbase = blockIdx.x * 16;      // 0 or 16 (B=32)
    const int nbase = blockIdx.y * 16;      // 0..1008
    const int r     = lane & 15;            // A-row within tile / B-col within tile
    const int half  = lane >> 4;            // 0: K={0,1}, 1: K={2,3}

    v8f c = {};
    const float* arow = h_t + (size_t)(mbase + r) * HSZ;
    for (int k0 = 0; k0 < HSZ; k0 += 4) {
        const int ka = k0 + 2 * half;
        v2f a, b;
        a.x = arow[ka];
        a.y = arow[ka + 1];
        b.x = W[(size_t)ka * HSZ + nbase + r];
        b.y = W[(size_t)(ka + 1) * HSZ + nbase + r];
        c = __builtin_amdgcn_wmma_f32_16x16x4_f32(
                /*neg_a=*/false, a, /*neg_b=*/false, b,
                /*c_mod=*/(short)0, c, /*reuse_a=*/false, /*reuse_b=*/false);
    }
#pragma unroll
    for (int j = 0; j < 8; ++j) {
        const int m = j + 8 * half;       // C/D row
        v[(size_t)(mbase + m) * HSZ + nbase + r] = c[j];
    }
}

// ---------------------------------------------------------------------------
// Kernel 2: per-(batch, S-chunk) flash partial.
// Each wave: 32 rows, online softmax state (m, l, acc) held in registers.
// Lane owns h = j*128 + lane*4 .. +3 for j = 0..7 (float4 slices).
// ---------------------------------------------------------------------------
__global__ __launch_bounds__(THREADS)
void cbsa_flash_partial(const float* __restrict__ cntx,
                        const float* __restrict__ vvec,
                        float* __restrict__ partials)
{
    __shared__ float s_v[HSZ];                 // 4 KB
    __shared__ float s_m[NWAVES];
    __shared__ float s_l[NWAVES];
    __shared__ float s_acc[NWAVES][HSZ];       // 32 KB

    const int b     = blockIdx.x;
    const int chunk = blockIdx.y;
    const int t     = threadIdx.x;
    const int wave  = t >> 5;
    const int lane  = t & 31;

    // Cooperative load of v[b,:] into LDS (256 threads x float4 = 1024 floats)
    ((float4*)s_v)[t] = ((const float4*)(vvec + (size_t)b * HSZ))[t];
    __syncthreads();

    // Per-lane register copy of v fragments
    float4 vf[8];
#pragma unroll
    for (int j = 0; j < 8; ++j) vf[j] = ((const float4*)s_v)[j * 32 + lane];

    float m = -INFINITY;
    float l = 0.0f;
    float4 acc[8];
#pragma unroll
    for (int j = 0; j < 8; ++j) acc[j] = make_float4(0.f, 0.f, 0.f, 0.f);

    const float* base =
        cntx + ((size_t)b * SSZ + (size_t)chunk * ROWS_PER_CHUNK) * HSZ;

    for (int rr = 0; rr < ROWS_PER_WAVE; ++rr) {
        const int row = wave + rr * NWAVES;  // adjacent waves touch adjacent rows
        const float4* rp = (const float4*)(base + (size_t)row * HSZ);

        float4 x[8];
#pragma unroll
        for (int j = 0; j < 8; ++j) x[j] = rp[j * 32 + lane];  // global_load_b128

        float d = 0.0f;
#pragma unroll
        for (int j = 0; j < 8; ++j) {
            d = fmaf(x[j].x, vf[j].x, d);
            d = fmaf(x[j].y, vf[j].y, d);
            d = fmaf(x[j].z, vf[j].z, d);
            d = fmaf(x[j].w, vf[j].w, d);
        }
        // wave32 reduction
#pragma unroll
        for (int off = 16; off > 0; off >>= 1) d += __shfl_xor(d, off, 32);

        // online softmax update (registers only)
        const float nm = fmaxf(m, d);
        const float p  = __expf(d - nm);
        const float sc = __expf(m - nm);   // m = -inf first iter -> 0
        l = l * sc + p;
#pragma unroll
        for (int j = 0; j < 8; ++j) {
            acc[j].x = fmaf(p, x[j].x, acc[j].x * sc);
            acc[j].y = fmaf(p, x[j].y, acc[j].y * sc);
            acc[j].z = fmaf(p, x[j].z, acc[j].z * sc);
            acc[j].w = fmaf(p, x[j].w, acc[j].w * sc);
        }
        m = nm;
    }

    // Publish wave partials to LDS
    if (lane == 0) { s_m[wave] = m; s_l[wave] = l; }
#pragma unroll
    for (int j = 0; j < 8; ++j) ((float4*)s_acc[wave])[j * 32 + lane] = acc[j];
    __syncthreads();

    // Combine 8 wave partials -> one workgroup partial
    float M = s_m[0];
#pragma unroll
    for (int w = 1; w < NWAVES; ++w) M = fmaxf(M, s_m[w]);
    float wgt[NWAVES];
    float L = 0.0f;
#pragma unroll
    for (int w = 0; w < NWAVES; ++w) {
        wgt[w] = __expf(s_m[w] - M);
        L = fmaf(s_l[w], wgt[w], L);
    }

    float* out = partials + ((size_t)b * NCHUNK + chunk) * PSTRIDE;
    if (t == 0) { out[0] = M; out[1] = L; }

    float4 s = make_float4(0.f, 0.f, 0.f, 0.f);
#pragma unroll
    for (int w = 0; w < NWAVES; ++w) {
        const float4 a = ((const float4*)s_acc[w])[t];
        s.x = fmaf(wgt[w], a.x, s.x);
        s.y = fmaf(wgt[w], a.y, s.y);
        s.z = fmaf(wgt[w], a.z, s.z);
        s.w = fmaf(wgt[w], a.w, s.w);
    }
    ((float4*)(out + 8))[t] = s;   // 16B-aligned: PSTRIDE*4 and +32B are multiples of 16
}

// ---------------------------------------------------------------------------
// Kernel 3: combine NCHUNK partials per batch, apply alpha/beta.
// out[b,h] = alpha*h_t[b,h] + beta * (sum_c acc_c[h]*e^{M_c-M}) / L
// ---------------------------------------------------------------------------
__global__ __launch_bounds__(THREADS)
void cbsa_finalize(const float* __restrict__ h_t,
                   const float* __restrict__ partials,
                   const float* __restrict__ alpha,
                   const float* __restrict__ beta,
                   float* __restrict__ out)
{
    const int b = blockIdx.x;
    const int t = threadIdx.x;

    const float* pb = partials + (size_t)b * NCHUNK * PSTRIDE;

    float M = -INFINITY;
#pragma unroll
    for (int c = 0; c < NCHUNK; ++c) M = fmaxf(M, pb[c * PSTRIDE]);

    float wgt[NCHUNK];
    float L = 0.0f;
#pragma unroll
    for (int c = 0; c < NCHUNK; ++c) {
        wgt[c] = __expf(pb[c * PSTRIDE] - M);
        L = fmaf(pb[c * PSTRIDE + 1], wgt[c], L);
    }

    float4 s = make_float4(0.f, 0.f, 0.f, 0.f);
#pragma unroll
    for (int c = 0; c < NCHUNK; ++c) {
        const float4 a = ((const float4*)(pb + c * PSTRIDE + 8))[t];
        s.x = fmaf(wgt[c], a.x, s.x);
        s.y = fmaf(wgt[c], a.y, s.y);
        s.z = fmaf(wgt[c], a.z, s.z);
        s.w = fmaf(wgt[c], a.w, s.w);
    }

    const float al  = alpha[0];
    const float inv = beta[0] / L;
    const float4 h  = ((const float4*)(h_t + (size_t)b * HSZ))[t];
    float4 o;
    o.x = fmaf(al, h.x, inv * s.x);
    o.y = fmaf(al, h.y, inv * s.y);
    o.z = fmaf(al, h.z, inv * s.z);
    o.w = fmaf(al, h.w, inv * s.w);
    ((float4*)(out + (size_t)b * HSZ))[t] = o;
}

// ---------------------------------------------------------------------------
extern "C" void kernel_launch(void* const* d_in, const int* in_sizes, int n_in,
                              void* d_out, int out_size, void* d_ws, size_t ws_size,
                              hipStream_t stream)
{
    const float* h_t   = (const float*)d_in[0];  // [B,H]
    const float* cntx  = (const float*)d_in[1];  // [B,S,H]
    const float* W     = (const float*)d_in[2];  // [H,H]
    const float* alpha = (const float*)d_in[3];  // [1]
    const float* beta  = (const float*)d_in[4];  // [1]
    float* out = (float*)d_out;                  // [B,H]

    // Workspace layout (floats): v[B*H], then partials[B*NCHUNK*PSTRIDE]
    float* ws_v    = (float*)d_ws;
    float* ws_part = ws_v + (size_t)BSZ * HSZ;

    // 1) v = h_t @ W  (WMMA fp32)
    cbsa_htW_wmma<<<dim3(BSZ / 16, HSZ / 16), 32, 0, stream>>>(h_t, W, ws_v);

    // 2) flash partials over cntx (single streaming pass over 268 MB)
    cbsa_flash_partial<<<dim3(BSZ, NCHUNK), THREADS, 0, stream>>>(cntx, ws_v, ws_part);

    // 3) combine + alpha/beta
    cbsa_finalize<<<dim3(BSZ), THREADS, 0, stream>>>(h_t, ws_part, alpha, beta, out);
}